// GaussianKDE_63385127354387
// MI455X (gfx1250) — compile-verified
//
#include <hip/hip_runtime.h>

// Weighted Gaussian KDE on gfx1250:
//   out[l] = sum_n w[n] * exp(-||x[l]-data[n]||^2 / SIGMA)
// Strategy: augmented rank-4 GEMM via V_WMMA_F32_16X16X4_F32 produces
//   D[m][n] = s*dist2(m,n) + log2(w[m]),  s = -1/(SIGMA*ln2)
// so the inner loop is just v_exp_f32 + v_add_f32 per pair.

typedef __attribute__((ext_vector_type(2))) float v2f;
typedef __attribute__((ext_vector_type(8))) float v8f;

#define KDE_SIGMA 3.0f
// s = -1/(SIGMA*ln2)
#define KDE_S (-0.4808983469629878f)
// 1/s = -SIGMA*ln2
#define KDE_INV_S (-2.0794415416798357f)

// ---------------------------------------------------------------------------
// Prep: aug[n][0..3] = ( d0, d1, |d|^2 + log2(w)/s, 1 )
// ---------------------------------------------------------------------------
__global__ void __launch_bounds__(256)
kde_prep_kernel(const float* __restrict__ data, const float* __restrict__ w,
                float* __restrict__ aug, int N) {
  int n = blockIdx.x * blockDim.x + threadIdx.x;
  if (n >= N) return;
  float d0 = data[2 * n + 0];
  float d1 = data[2 * n + 1];
  float lw = __builtin_amdgcn_logf(w[n]);          // v_log_f32 == log2(x)
  aug[4 * n + 0] = d0;
  aug[4 * n + 1] = d1;
  aug[4 * n + 2] = d0 * d0 + d1 * d1 + lw * KDE_INV_S;
  aug[4 * n + 3] = 1.0f;
}

// ---------------------------------------------------------------------------
// Main: each wave handles 16 locations, sweeps all N points in 16-point tiles.
// ---------------------------------------------------------------------------
__global__ void __launch_bounds__(256)
kde_main_kernel(const float* __restrict__ x, const float* __restrict__ aug,
                float* __restrict__ out, int ntiles) {
  const int lane = threadIdx.x & 31;
  const int wave = blockIdx.x * (blockDim.x >> 5) + (threadIdx.x >> 5);
  const int j = lane & 15;                 // location column within tile
  const int loc = wave * 16 + j;
  const bool hi = lane >= 16;

  // Build B (4x16, x-augmented) once; lives in 2 VGPRs for the whole kernel.
  // lanes 0-15 : VGPR0 = B[0][j] = -2*s*x0 , VGPR1 = B[1][j] = -2*s*x1
  // lanes 16-31: VGPR0 = B[2][j] =  s      , VGPR1 = B[3][j] =  s*|x|^2
  float x0 = x[2 * loc + 0];
  float x1 = x[2 * loc + 1];
  v2f b;
  b.x = hi ? KDE_S : (-2.0f * KDE_S) * x0;
  b.y = hi ? KDE_S * (x0 * x0 + x1 * x1) : (-2.0f * KDE_S) * x1;

  // Per-lane A pointer (ISA 16x4 F32 A layout):
  // lanes 0-15 load (A[m][0],A[m][1]); lanes 16-31 load (A[m][2],A[m][3]); m=j.
  const float* ap = aug + (size_t)j * 4 + (hi ? 2 : 0);

  v8f acc = {};
#pragma unroll 4
  for (int t = 0; t < ntiles; ++t) {
    v2f a = *(const v2f*)(ap + (size_t)t * 64);  // global_load_b64, tile stride 256B
    v8f c = {};
    // D[m][n] = s*dist2 + log2(w[m]) over the 16x16 tile
    v8f d = __builtin_amdgcn_wmma_f32_16x16x4_f32(
        /*neg_a=*/false, a, /*neg_b=*/false, b,
        /*c_mod=*/(short)0, c, /*reuse_a=*/false, /*reuse_b=*/false);
#pragma unroll
    for (int v = 0; v < 8; ++v)
      acc[v] += __builtin_amdgcn_exp2f(d[v]);    // v_exp_f32 + v_add_f32
  }

  // Lane j holds points m=0..7 of each tile, lane j+16 holds m=8..15 (same loc).
  float tot = ((acc[0] + acc[1]) + (acc[2] + acc[3])) +
              ((acc[4] + acc[5]) + (acc[6] + acc[7]));
  tot += __shfl_xor(tot, 16, 32);
  if (!hi) out[loc] = tot;
}

// ---------------------------------------------------------------------------
extern "C" void kernel_launch(void* const* d_in, const int* in_sizes, int n_in,
                              void* d_out, int out_size, void* d_ws, size_t ws_size,
                              hipStream_t stream) {
  const float* x    = (const float*)d_in[0];  // (B, L, 2) f32
  const float* data = (const float*)d_in[1];  // (N, 2)    f32
  const float* w    = (const float*)d_in[2];  // (N,)      f32
  float* out = (float*)d_out;                 // (B, L)    f32

  const int N    = in_sizes[2];               // 16384
  const int nloc = out_size;                  // B*L = 131072
  float* aug = (float*)d_ws;                  // N*4 floats = 256 KB scratch

  kde_prep_kernel<<<(N + 255) / 256, 256, 0, stream>>>(data, w, aug, N);

  const int waves  = nloc / 16;               // 16 locations per wave
  const int blocks = waves / 8;               // 8 waves (256 thr) per block
  kde_main_kernel<<<blocks, 256, 0, stream>>>(x, aug, out, N / 16);
}